// MK_MMDLoss_55473797595639
// MI455X (gfx1250) — compile-verified
//
#include <hip/hip_runtime.h>

typedef float v2f __attribute__((ext_vector_type(2)));
typedef float v8f __attribute__((ext_vector_type(8)));

#define NROWS 4096
#define CFEAT 64

// ---------------------------------------------------------------------------
// ws layout (floats unless noted):
//   xf  : [0, 4096*64)              flattened x, row-major (n, c)
//   yf  : [4096*64, 2*4096*64)      flattened y
//   a2x : 4096 row squared-norms of xf
//   a2y : 4096 row squared-norms of yf
//   acc : 3 doubles (xx, yy, xy partial sums)
// total ~2.03 MiB
// ---------------------------------------------------------------------------

__global__ void init_acc_kernel(double* __restrict__ acc) {
  if (threadIdx.x < 3) acc[threadIdx.x] = 0.0;
}

// (B,C,H,W) -> (B*H*W, C) row-major. n = b*1024 + hw, feature c.
__global__ void pack_kernel(const float* __restrict__ x, const float* __restrict__ y,
                            float* __restrict__ xf, float* __restrict__ yf) {
  int idx = blockIdx.x * blockDim.x + threadIdx.x;   // 2 * 262144 threads
  int t   = idx >> 18;
  int r   = idx & 262143;
  int c   = r >> 12;          // [0,64)
  int n   = r & 4095;         // [0,4096) -- consecutive threads: coalesced reads
  int b   = n >> 10;
  int hw  = n & 1023;
  const float* src = t ? y : x;
  float*       dst = t ? yf : xf;
  dst[n * CFEAT + c] = src[(b * CFEAT + c) * 1024 + hw];
}

__global__ void norm_kernel(const float* __restrict__ xf, const float* __restrict__ yf,
                            float* __restrict__ a2x, float* __restrict__ a2y) {
  int idx = blockIdx.x * blockDim.x + threadIdx.x;   // 8192 threads
  int t   = idx >> 12;
  int n   = idx & 4095;
  const float4* row = (const float4*)((t ? yf : xf) + n * CFEAT);
  float s = 0.0f;
#pragma unroll
  for (int i = 0; i < 16; ++i) {
    float4 v = row[i];
    s += v.x * v.x + v.y * v.y + v.z * v.z + v.w * v.w;
  }
  (t ? a2y : a2x)[n] = s;
}

// Main kernel: grid = (512, 3), block = 256 (8 waves of 32).
// blockIdx.y selects the pair: 0 -> (xf,xf), 1 -> (yf,yf), 2 -> (xf,yf).
// Each wave owns one 16-row tile of A and sweeps 16 column tiles of B,
// doing D = A*B^T via V_WMMA_F32_16X16X4_F32 (K=64 in 16 steps of K=4),
// then the 5-sigma Gaussian kernel sum on the 16x16 distance tile.
__launch_bounds__(256)
__global__ void mmd_wmma_kernel(const float* __restrict__ xf, const float* __restrict__ yf,
                                const float* __restrict__ a2x, const float* __restrict__ a2y,
                                double* __restrict__ acc) {
  const int pair = blockIdx.y;
  const float* Am = (pair == 1) ? yf  : xf;
  const float* Bm = (pair == 0) ? xf  : yf;
  const float* nA = (pair == 1) ? a2y : a2x;
  const float* nB = (pair == 0) ? a2x : a2y;

  const int wave = threadIdx.x >> 5;
  const int lane = threadIdx.x & 31;
  const int half = lane >> 4;     // 0: lanes 0-15, 1: lanes 16-31
  const int l15  = lane & 15;

  // 16 strips of 16 column tiles; 32 blocks per strip; 8 waves/block -> i tile
  const int strip = blockIdx.x >> 5;                    // [0,16)
  const int itile = ((blockIdx.x & 31) << 3) + wave;    // [0,256)

  // --- Preload A fragments for this wave's 16-row strip (kept in VGPRs) ---
  // A 16x4 f32 layout: lanes 0-15 row M=l15 hold K={4k,4k+1}; lanes 16-31 K={4k+2,4k+3}
  const int rowA = itile * 16 + l15;
  v2f afrag[16];
#pragma unroll
  for (int k = 0; k < 16; ++k)
    afrag[k] = *(const v2f*)(Am + rowA * CFEAT + k * 4 + half * 2);

  // C/D layout: VGPR r <-> row M = r + 8*half, col N = l15
  float a2row[8];
#pragma unroll
  for (int r = 0; r < 8; ++r)
    a2row[r] = nA[itile * 16 + 8 * half + r];

  const float betas[5] = {50.0f, 5.0f, 0.5f, 0.05f, 0.005f};  // 1/(2*sigma)
  float lsum = 0.0f;

  for (int jt = 0; jt < 16; ++jt) {
    const int jtile = strip * 16 + jt;
    const int rowB  = jtile * 16 + l15;      // output column index
    const float b2v = nB[rowB];

    v8f cacc = {0.f, 0.f, 0.f, 0.f, 0.f, 0.f, 0.f, 0.f};
#pragma unroll
    for (int k = 0; k < 16; ++k) {
      // B 4x16 f32 layout mirrors A with N=l15: lanes 0-15 K={4k,4k+1}, lanes 16-31 K={4k+2,4k+3}
      v2f bfrag = *(const v2f*)(Bm + rowB * CFEAT + k * 4 + half * 2);
      cacc = __builtin_amdgcn_wmma_f32_16x16x4_f32(
          /*neg_a=*/false, afrag[k], /*neg_b=*/false, bfrag,
          /*c_mod=*/(short)0, cacc, /*reuse_a=*/false, /*reuse_b=*/false);
    }

#pragma unroll
    for (int r = 0; r < 8; ++r) {
      float d2 = a2row[r] + b2v - 2.0f * cacc[r];
      d2 = fmaxf(d2, 0.0f);
#pragma unroll
      for (int s = 0; s < 5; ++s)
        lsum += __expf(-betas[s] * d2);
    }
  }

  // wave32 shuffle reduction, then one f64 atomic per wave
#pragma unroll
  for (int off = 16; off > 0; off >>= 1)
    lsum += __shfl_down(lsum, off, 32);
  if (lane == 0)
    atomicAdd(&acc[pair], (double)lsum);
}

__global__ void finalize_kernel(const double* __restrict__ acc, float* __restrict__ out) {
  if (threadIdx.x == 0) {
    const double inv = 1.0 / (4096.0 * 4096.0);
    out[0] = (float)((acc[0] + acc[1] - 2.0 * acc[2]) * inv);
  }
}

extern "C" void kernel_launch(void* const* d_in, const int* in_sizes, int n_in,
                              void* d_out, int out_size, void* d_ws, size_t ws_size,
                              hipStream_t stream) {
  const float* x = (const float*)d_in[0];
  const float* y = (const float*)d_in[1];
  float* out = (float*)d_out;

  float* xf  = (float*)d_ws;
  float* yf  = xf + NROWS * CFEAT;
  float* a2x = yf + NROWS * CFEAT;
  float* a2y = a2x + NROWS;
  double* acc = (double*)(a2y + NROWS);   // byte offset 2,129,920 -> 8B aligned

  init_acc_kernel<<<1, 32, 0, stream>>>(acc);
  pack_kernel<<<2048, 256, 0, stream>>>(x, y, xf, yf);
  norm_kernel<<<32, 256, 0, stream>>>(xf, yf, a2x, a2y);
  dim3 grid(512, 3);
  mmd_wmma_kernel<<<grid, 256, 0, stream>>>(xf, yf, a2x, a2y, acc);
  finalize_kernel<<<1, 32, 0, stream>>>(acc, out);
}